// TopK_17360257810768
// MI455X (gfx1250) — compile-verified
//
#include <hip/hip_runtime.h>
#include <math.h>

// MI455X (gfx1250) TopK-gather pipeline.
// Bandwidth-bound: ~213 MB total traffic -> ~9.2 us floor at 23.3 TB/s.
// GEMV scoring runs on V_WMMA_F32_16X16X4_F32 (16-row tiles per wave,
// scorer chunk broadcast across all 16 B columns; column 0 of D = dots).
// Round-1 fix: unconditional clamped loads in the streaming loop so A/B
// both lower to straight-line global_load_b64 (no exec-mask branches).

typedef float v2f __attribute__((ext_vector_type(2)));
typedef float v8f __attribute__((ext_vector_type(8)));

#define KSEL 4096
#define FDIM 256

__device__ __forceinline__ unsigned keyEncode(float s) {
    unsigned b = __float_as_uint(s);
    return (b & 0x80000000u) ? ~b : (b | 0x80000000u); // monotone: bigger float -> bigger uint
}
__device__ __forceinline__ float keyDecode(unsigned u) {
    unsigned b = (u & 0x80000000u) ? (u ^ 0x80000000u) : ~u;
    return __uint_as_float(b);
}

// ---------------- init: zero per-call scratch state ----------------
__global__ void init_kernel(unsigned* hist, unsigned* state) {
    int t = threadIdx.x;
    if (t < 1024) hist[t] = 0u;   // 4 passes x 256 bins
    if (t < 16)   state[t] = 0u;
    if (t == 0)   state[1] = KSEL; // remaining
}

// ---------------- scorer normalization ----------------
__global__ void norm_kernel(const float* __restrict__ scorer, float* __restrict__ wscaled) {
    __shared__ float red[FDIM];
    int t = threadIdx.x;
    float v = scorer[t];
    red[t] = v * v;
    __syncthreads();
    for (int s = FDIM / 2; s > 0; s >>= 1) {
        if (t < s) red[t] += red[t + s];
        __syncthreads();
    }
    float scale = rsqrtf(red[0]);
    wscaled[t] = v * scale;
}

// ---------------- WMMA GEMV scoring ----------------
// Wave handles 16 rows. A = 16x4 f32 tile of node_embs (ISA layout:
// lanes 0-15 hold the K=0,1 pair; lanes 16-31 hold the K=2,3 pair).
// B = 4x16 with the normalized scorer chunk replicated in every column,
// so every D column equals the 16 dot products; lanes 0/16 read column 0.
// Loads are UNCONDITIONAL from a clamped row: OOB lanes compute garbage
// that is never written (write-out is range-guarded), keeping the stream
// loop branch-free and EXEC all-1s for WMMA.
__global__ void __launch_bounds__(256) score_kernel(const float* __restrict__ emb,
                                                    const float* __restrict__ w,
                                                    unsigned* __restrict__ keys, int N) {
    int lane = threadIdx.x & 31;
    int wave = threadIdx.x >> 5;
    int rowBase = (blockIdx.x * 8 + wave) * 16;
    int row  = rowBase + (lane & 15);
    int rowC = (row < N) ? row : (N - 1);
    const v2f* __restrict__ rp = (const v2f*)(emb + (size_t)rowC * FDIM);
    const v2f* __restrict__ wp = (const v2f*)w;
    int kp = lane >> 4;                 // pair offset: 0 (lanes 0-15) / 1 (lanes 16-31)

    // Prefetch the next wave-tile's row start to keep the HBM stream ahead.
    {
        int nrow = row + 16;
        if (nrow < N) __builtin_prefetch(emb + (size_t)nrow * FDIM, 0, 1);
    }

    v8f acc = {0.f, 0.f, 0.f, 0.f, 0.f, 0.f, 0.f, 0.f};
    #pragma unroll 8
    for (int p = 0; p < FDIM / 2; p += 2) {   // p indexes float-pairs; +2 pairs = K+4 per WMMA
        v2f a = rp[p + kp];                   // global_load_b64, branch-free
        v2f b = wp[p + kp];                   // global_load_b64, L0-resident
        acc = __builtin_amdgcn_wmma_f32_16x16x4_f32(
            /*neg_a=*/false, a, /*neg_b=*/false, b,
            /*c_mod=*/(short)0, acc, /*reuse_a=*/false, /*reuse_b=*/false);
    }

    // C/D layout: lane 0 VGPR r -> D[r][0]; lane 16 VGPR r -> D[8+r][0]
    if ((lane & 15) == 0) {
        int mBase = kp * 8;                   // 0 for lane 0, 8 for lane 16
        int obase = rowBase + mBase;
        if (obase + 8 <= N) {
            uint4 lo, hi;
            lo.x = keyEncode(acc[0]); lo.y = keyEncode(acc[1]);
            lo.z = keyEncode(acc[2]); lo.w = keyEncode(acc[3]);
            hi.x = keyEncode(acc[4]); hi.y = keyEncode(acc[5]);
            hi.z = keyEncode(acc[6]); hi.w = keyEncode(acc[7]);
            *(uint4*)(keys + obase)     = lo;  // global_store_b128
            *(uint4*)(keys + obase + 4) = hi;  // global_store_b128
        } else {
            #pragma unroll
            for (int r = 0; r < 8; ++r)
                if (obase + r < N) keys[obase + r] = keyEncode(acc[r]);
        }
    }
}

// ---------------- radix-select histogram pass ----------------
__global__ void hist_kernel(const unsigned* __restrict__ keys, unsigned* __restrict__ hist,
                            const unsigned* __restrict__ state, int N, unsigned mask, int shift) {
    __shared__ unsigned lh[256];
    int t = threadIdx.x;
    lh[t] = 0u;
    __syncthreads();
    unsigned prefix = state[0] & mask;
    int i = blockIdx.x * 256 + t;
    if (i < N) {
        unsigned u = keys[i];
        if ((u & mask) == prefix) atomicAdd(&lh[(u >> shift) & 255u], 1u);
    }
    __syncthreads();
    if (lh[t]) atomicAdd(&hist[t], lh[t]);
}

// ---------------- radix-select scan (tiny, serial) ----------------
__global__ void scan_kernel(const unsigned* __restrict__ hist, unsigned* state,
                            int shift, int finalPass) {
    if (threadIdx.x == 0) {
        unsigned remaining = state[1];
        int bin = 0;
        for (int b = 255; b >= 0; --b) {
            unsigned c = hist[b];
            if (remaining > c) { remaining -= c; }
            else { bin = b; break; }
        }
        state[0] |= ((unsigned)bin << shift);
        state[1] = remaining;
        if (finalPass) {
            state[2] = KSEL - remaining; // count strictly greater than T
            state[3] = remaining;        // ties needed at T
        }
    }
}

// ---------------- partition candidates ----------------
__global__ void select_kernel(const unsigned* __restrict__ keys, unsigned* state,
                              unsigned* __restrict__ candKey, unsigned* __restrict__ candIdx,
                              int N) {
    int i = blockIdx.x * 256 + threadIdx.x;
    if (i >= N) return;
    unsigned u = keys[i];
    unsigned T = state[0];
    if (u > T) {
        unsigned p = atomicAdd(&state[4], 1u);
        candKey[p] = u; candIdx[p] = (unsigned)i;
    } else if (u == T) {
        unsigned e = atomicAdd(&state[5], 1u);
        if (e < state[3]) {
            unsigned p = state[2] + e;
            candKey[p] = u; candIdx[p] = (unsigned)i;
        }
    }
}

// ---------------- bitonic sort of 4096 candidates (desc key, asc idx) ----------------
__global__ void __launch_bounds__(1024) sort_kernel(unsigned* __restrict__ ck,
                                                    unsigned* __restrict__ ci) {
    __shared__ unsigned sk[KSEL];
    __shared__ unsigned si[KSEL];
    int t = threadIdx.x;
    for (int i = t; i < KSEL; i += 1024) { sk[i] = ck[i]; si[i] = ci[i]; }
    __syncthreads();
    for (int size = 2; size <= KSEL; size <<= 1) {
        for (int stride = size >> 1; stride > 0; stride >>= 1) {
            for (int i = t; i < KSEL; i += 1024) {
                int j = i ^ stride;
                if (j > i) {
                    unsigned ki = sk[i], kj = sk[j], ii = si[i], ij = si[j];
                    // "i ranks after j" under (key desc, idx asc)
                    bool iAfter = (ki < kj) || (ki == kj && ii > ij);
                    bool dirDesc = ((i & size) == 0);
                    bool doSwap = dirDesc ? iAfter : !iAfter;
                    if (doSwap) { sk[i] = kj; sk[j] = ki; si[i] = ij; si[j] = ii; }
                }
            }
            __syncthreads();
        }
    }
    for (int i = t; i < KSEL; i += 1024) { ck[i] = sk[i]; ci[i] = si[i]; }
}

// ---------------- gather + tanh-scale + transpose ----------------
// Block handles 64 output columns. Rows staged in LDS with stride 257
// (conflict-free transpose reads); stores are 128B coalesced along k.
__global__ void __launch_bounds__(256) gather_kernel(const float* __restrict__ emb,
                                                     const unsigned* __restrict__ candKey,
                                                     const unsigned* __restrict__ candIdx,
                                                     float* __restrict__ out) {
    __shared__ float tile[64 * 257];
    __shared__ float tv[64];
    int t = threadIdx.x;
    int jbase = blockIdx.x * 64;
    if (t < 64) tv[t] = tanhf(keyDecode(candKey[jbase + t]));
    __syncthreads();
    for (int r = 0; r < 64; ++r) {
        unsigned idx = candIdx[jbase + r];
        tile[r * 257 + t] = emb[(size_t)idx * FDIM + t]; // coalesced row load
    }
    __syncthreads();
    int jj = t & 63;
    int f0 = t >> 6; // 0..3
    #pragma unroll
    for (int it = 0; it < 64; ++it) {
        int f = it * 4 + f0;
        out[(size_t)f * KSEL + jbase + jj] = tile[jj * 257 + f] * tv[jj];
    }
}

extern "C" void kernel_launch(void* const* d_in, const int* in_sizes, int n_in,
                              void* d_out, int out_size, void* d_ws, size_t ws_size,
                              hipStream_t stream) {
    const float* emb    = (const float*)d_in[0];
    const float* scorer = (const float*)d_in[1];
    (void)n_in; (void)out_size; (void)ws_size;
    int F = in_sizes[1];           // 256
    int N = in_sizes[0] / F;       // 200000

    // ws layout (all 16B-aligned):
    unsigned* keys    = (unsigned*)d_ws;                         // N keys
    size_t nAlign     = ((size_t)N + 63) & ~(size_t)63;
    float*    wscaled = (float*)(keys + nAlign);                 // 256 floats
    unsigned* hist    = (unsigned*)(wscaled + FDIM);             // 4*256
    unsigned* state   = hist + 1024;                             // 16
    unsigned* candKey = state + 32;                              // 4096
    unsigned* candIdx = candKey + KSEL;                          // 4096

    init_kernel<<<1, 1024, 0, stream>>>(hist, state);
    norm_kernel<<<1, FDIM, 0, stream>>>(scorer, wscaled);

    int scoreBlocks = (N + 127) / 128;   // 8 waves/block, 16 rows/wave
    score_kernel<<<scoreBlocks, 256, 0, stream>>>(emb, wscaled, keys, N);

    int nb = (N + 255) / 256;
    for (int p = 0; p < 4; ++p) {
        int shift = 24 - 8 * p;
        unsigned mask = (p == 0) ? 0u : (0xFFFFFFFFu << (32 - 8 * p));
        hist_kernel<<<nb, 256, 0, stream>>>(keys, hist + p * 256, state, N, mask, shift);
        scan_kernel<<<1, 32, 0, stream>>>(hist + p * 256, state, shift, (p == 3) ? 1 : 0);
    }

    select_kernel<<<nb, 256, 0, stream>>>(keys, state, candKey, candIdx, N);
    sort_kernel<<<1, 1024, 0, stream>>>(candKey, candIdx);
    gather_kernel<<<KSEL / 64, 256, 0, stream>>>(emb, candKey, candIdx, (float*)d_out);
}